// SurfEval_30846455119882
// MI455X (gfx1250) — compile-verified
//
#include <hip/hip_runtime.h>

// NURBS surface eval: B=64 batches, 64x64 homogeneous control grid, cubic x
// cubic, 512x512 samples, perspective divide.
//
// Roofline: output = 64*512*512*3 f32 = 192 MB of stores -> ~8.5us floor at
// 23.3 TB/s HBM; FLOPs (~1G with the separable factorization) are noise.
// Therefore: fp32 end-to-end, f32 WMMA (V_WMMA_F32_16X16X4_F32) for the
// v-contraction, async global->LDS staging of the control-point window, fast
// rcp+NR for the divide (avoids the ~12-instr IEEE div sequence x 16.7M).
//
// Window trick (both dims): spans advance <=2 per 16 consecutive samples
// (knot spacing 1/60 vs sample spacing 1/511), so the 4-wide gather windows
// of a 16-sample tile fit in a dense 8-wide window [c0, c0+8). This makes
// each 16x16 output tile an exact dense 16x8 @ 8x16 GEMM = two K=4 f32 WMMAs.

typedef __attribute__((ext_vector_type(2))) float v2f;
typedef __attribute__((ext_vector_type(8))) float v8f;

#define NB     64
#define NM     64
#define NN     64
#define OUTDIM 512
#define NVT    32   // 512/16 v-tiles
#define NUT    32   // 512/16 u-tiles

// ---------------------------------------------------------------------------
// Per-v-tile compacted dense basis: Bmat[vt][k][vl] (k<8, vl<16)
//   Bmat[vt][k][vl] = sum_r Nv[v,r] * (iv[v,r] == c0[vt]+k),  v = vt*16+vl
// iv rows are sorted and spans are monotonic, so c0[vt] = iv[vt*16][0].
// ---------------------------------------------------------------------------
__global__ void build_vtile_kernel(const float* __restrict__ Nv,
                                   const int*   __restrict__ iv,
                                   float*       __restrict__ Bmat,
                                   int*         __restrict__ c0arr) {
    const int vt = blockIdx.x;      // 0..31
    const int vl = threadIdx.x;     // 0..15
    const int v  = vt * 16 + vl;
    const int c0 = iv[(vt * 16) * 4 + 0];
    if (vl == 0) c0arr[vt] = c0;

    float col[8];
#pragma unroll
    for (int k = 0; k < 8; ++k) col[k] = 0.0f;
#pragma unroll
    for (int r = 0; r < 4; ++r) {
        int   k = iv[v * 4 + r] - c0;
        float w = Nv[v * 4 + r];
        if (k >= 0 && k < 8) col[k] += w;
    }
#pragma unroll
    for (int k = 0; k < 8; ++k)
        Bmat[(vt * 8 + k) * 16 + vl] = col[k];
}

// ---------------------------------------------------------------------------
// Fused kernel: one workgroup per (b, u-tile); 256 threads = 8 wave32.
//  0) async-copy the 8-row ctrl window (8 KB) into LDS (ASYNCcnt path)
//  1) temp[16u][64n][4c] = sum_l Nu[u,l]*ctrlWin[iu[u,l]-c0u][n][:]   (LDS)
//  2) per v-tile, per component: 16x16 tile via two chained K=4 f32 WMMAs,
//     then rcp+NR perspective divide and non-temporal 12B/point stores.
// ---------------------------------------------------------------------------
__global__ __launch_bounds__(256) void surfeval_kernel(
    const float4* __restrict__ ctrl,   // (64,64,64,4)
    const float*  __restrict__ Nu,     // (512,4)
    const int*    __restrict__ iu,     // (512,4)
    const float*  __restrict__ Bmat,   // (32,8,16)
    const int*    __restrict__ c0arr,  // (32)
    float*        __restrict__ out) {  // (64,512,512,3)

    const int b  = blockIdx.x >> 5;
    const int ut = blockIdx.x & 31;
    const int t  = threadIdx.x;

    __shared__ float4 ctrlWin[8][64];       // 8 KB ctrl row window
    __shared__ float  temp[16][65][4];      // u-stride 260 dwords: bank-clean

    // ---- Stage 0: async global->LDS copy of the u-window ------------------
    const int c0u = iu[(ut * 16) * 4 + 0];  // min row of this u-tile's window
    {
        const uint32_t ldsBase = (uint32_t)(uintptr_t)(void*)&ctrlWin[0][0];
#pragma unroll
        for (int q = 0; q < 2; ++q) {
            const int chunk = t + q * 256;          // 0..511 = 8 rows x 64 n
            const int rr    = chunk >> 6;
            const int n     = chunk & 63;
            int row = c0u + rr;
            row = row > 63 ? 63 : row;              // clamped dups never read
            const float4* gp = ctrl + ((size_t)(b * NM + row) * NN + n);
            const uint32_t lds = ldsBase + (uint32_t)chunk * 16u;
            asm volatile("global_load_async_to_lds_b128 %0, %1, off"
                         :: "v"(lds), "v"((unsigned long long)(uintptr_t)gp)
                         : "memory");
        }
    }

    // Independent work while the async copy flies: per-u weights.
    const int ulw = t >> 4;                 // this thread's u row (0..15)
    const int nb  = t & 15;                 // base column
    const int ug  = ut * 16 + ulw;
    int   kk[4];
    float ww[4];
#pragma unroll
    for (int l = 0; l < 4; ++l) {
        kk[l] = iu[ug * 4 + l] - c0u;       // in [0,7] by span monotonicity
        ww[l] = Nu[ug * 4 + l];
    }

    asm volatile("s_wait_asynccnt 0" ::: "memory");
    __syncthreads();

    // ---- Stage 1: gather-FMA from the LDS window --------------------------
#pragma unroll
    for (int q = 0; q < 4; ++q) {
        const int n = nb + q * 16;
        float ax = 0.f, ay = 0.f, az = 0.f, aw = 0.f;
#pragma unroll
        for (int l = 0; l < 4; ++l) {
            const float4 p = ctrlWin[kk[l]][n];
            const float  w = ww[l];
            ax = fmaf(w, p.x, ax);
            ay = fmaf(w, p.y, ay);
            az = fmaf(w, p.z, az);
            aw = fmaf(w, p.w, aw);
        }
        temp[ulw][n][0] = ax;
        temp[ulw][n][1] = ay;
        temp[ulw][n][2] = az;
        temp[ulw][n][3] = aw;
    }
    __syncthreads();

    // ---- Stage 2: WMMA over v-tiles ---------------------------------------
    const int wave = t >> 5;        // 0..7
    const int lane = t & 31;
    const int half = lane >> 4;     // 0: lanes 0-15, 1: lanes 16-31
    const int mn   = lane & 15;     // = M (A/D row) = N (B/D col)
    const int kk0  = half * 2;      // frag K pair base per ISA layout

    for (int vt = wave; vt < NVT; vt += 8) {
        const int c0 = c0arr[vt];

        // B frags (4x16: lane N holds K0/K1, lane N+16 holds K2/K3)
        v2f bf0, bf1;
        bf0.x = Bmat[(vt * 8 + kk0    ) * 16 + mn];
        bf0.y = Bmat[(vt * 8 + kk0 + 1) * 16 + mn];
        bf1.x = Bmat[(vt * 8 + 4 + kk0    ) * 16 + mn];
        bf1.y = Bmat[(vt * 8 + 4 + kk0 + 1) * 16 + mn];

        v8f d[4];
#pragma unroll
        for (int c = 0; c < 4; ++c) {
            // A frags (16x4: lane M holds K0/K1, lane M+16 holds K2/K3).
            // Clamp: columns with c0+k > 63 have Bmat rows == 0, clamping
            // just keeps the unused A operand finite.
            int n0 = c0 + kk0;
            int i0 = n0     > 63 ? 63 : n0;
            int i1 = n0 + 1 > 63 ? 63 : n0 + 1;
            int n4 = c0 + 4 + kk0;
            int i2 = n4     > 63 ? 63 : n4;
            int i3 = n4 + 1 > 63 ? 63 : n4 + 1;
            v2f af0, af1;
            af0.x = temp[mn][i0][c];
            af0.y = temp[mn][i1][c];
            af1.x = temp[mn][i2][c];
            af1.y = temp[mn][i3][c];

            v8f acc = {};
            acc = __builtin_amdgcn_wmma_f32_16x16x4_f32(
                      false, af0, false, bf0, (short)0, acc, false, false);
            acc = __builtin_amdgcn_wmma_f32_16x16x4_f32(
                      false, af1, false, bf1, (short)0, acc, false, false);
            d[c] = acc;
        }

        // D layout: VGPR j -> row M = j (lanes 0-15) / j+8 (lanes 16-31).
        const int v = vt * 16 + mn;
#pragma unroll
        for (int j = 0; j < 8; ++j) {
            const int u = ut * 16 + j + half * 8;
            // Fast reciprocal + one Newton step (~ulp^2): w is a convex
            // combination of uniform(0.5,1.5) weights -> well conditioned.
            const float w = d[3][j];
            float r = __builtin_amdgcn_rcpf(w);
            r = fmaf(fmaf(-w, r, 1.0f), r, r);
            const size_t base = (((size_t)b * OUTDIM + u) * OUTDIM + v) * 3;
            __builtin_nontemporal_store(d[0][j] * r, out + base + 0);
            __builtin_nontemporal_store(d[1][j] * r, out + base + 1);
            __builtin_nontemporal_store(d[2][j] * r, out + base + 2);
        }
    }
}

// ---------------------------------------------------------------------------
extern "C" void kernel_launch(void* const* d_in, const int* in_sizes, int n_in,
                              void* d_out, int out_size, void* d_ws, size_t ws_size,
                              hipStream_t stream) {
    const float4* ctrl = (const float4*)d_in[0];  // (64,64,64,4) f32
    const float*  Nu   = (const float*) d_in[1];  // (512,4) f32
    const float*  Nv   = (const float*) d_in[2];  // (512,4) f32
    const int*    iu   = (const int*)   d_in[3];  // (512,4) int
    const int*    iv   = (const int*)   d_in[4];  // (512,4) int
    float*        out  = (float*)d_out;           // (64,512,512,3) f32

    float* Bmat  = (float*)d_ws;                  // 32*8*16 f32 = 16 KB
    int*   c0arr = (int*)(Bmat + NVT * 8 * 16);   // +32 ints

    build_vtile_kernel<<<NVT, 16, 0, stream>>>(Nv, iv, Bmat, c0arr);

    const int nblocks = NB * NUT;                 // 2048 workgroups
    surfeval_kernel<<<nblocks, 256, 0, stream>>>(ctrl, Nu, iu, Bmat, c0arr, out);
}